// LaplacianLoss_62294205661545
// MI455X (gfx1250) — compile-verified
//
#include <hip/hip_runtime.h>

// ---------------- problem constants ----------------
#define VNUM   6890
#define NFACES 13776
#define BATCH  32
#define NCOLS  96              // BATCH * 3
#define NT     431             // ceil(VNUM/16) vertex tiles
#define VPAD   6896            // NT*16
#define WROW   216             // ceil(VNUM/32) u32 words per adjacency row
#define BMW    14              // ceil(NT/32) words per block-bitmap row
#define EPS    1e-8f

// ---------------- workspace layout (u32 words) ----------------
static constexpr size_t ADJ_OFF  = 0;
static constexpr size_t ADJ_W    = (size_t)VPAD * WROW;      // 1,489,536 words
static constexpr size_t DEG_OFF  = ADJ_OFF + ADJ_W;
static constexpr size_t INV_OFF  = DEG_OFF + VPAD;
static constexpr size_t XT_OFF   = INV_OFF + VPAD;
static constexpr size_t XT_W     = (size_t)NCOLS * VPAD;     // 662,016 words (col-major)
static constexpr size_t BM_OFF   = XT_OFF + XT_W;
static constexpr size_t BM_W     = (size_t)NT * BMW;         // 6,034 words
static constexpr size_t PART_OFF = BM_OFF + BM_W;
static constexpr size_t TOTAL_W  = PART_OFF + 512;           // ~8.3 MB total

typedef __attribute__((ext_vector_type(2))) float v2f;
typedef __attribute__((ext_vector_type(8))) float v8f;

// ---------------- kernels ----------------

__global__ void k_zero(unsigned* __restrict__ w, long n) {
  long i = (long)blockIdx.x * blockDim.x + threadIdx.x;
  long stride = (long)gridDim.x * blockDim.x;
  for (; i < n; i += stride) w[i] = 0u;
}

// Set adjacency bits for all 6 directed pairs of each face (atomicOr is
// order-independent -> deterministic). Self-pairs set the diagonal bit,
// which (matching the reference) counts toward deg but not toward the
// off-diagonal -1 entries.
__global__ void k_adj(const int* __restrict__ faces, unsigned* __restrict__ adj) {
  int f = blockIdx.x * blockDim.x + threadIdx.x;
  if (f >= NFACES) return;
  int i0 = faces[f * 3 + 0], i1 = faces[f * 3 + 1], i2 = faces[f * 3 + 2];
  auto setb = [&](int r, int c) {
    atomicOr(&adj[(size_t)r * WROW + (c >> 5)], 1u << (c & 31));
  };
  setb(i0, i1); setb(i1, i0);
  setb(i1, i2); setb(i2, i1);
  setb(i2, i0); setb(i0, i2);
}

// deg[v] = popcount of row v (includes diagonal bit, per reference semantics)
__global__ void k_deg(const unsigned* __restrict__ adj,
                      float* __restrict__ degf, float* __restrict__ invf) {
  int wave = threadIdx.x >> 5, lane = threadIdx.x & 31;
  int v = blockIdx.x * 8 + wave;
  if (v >= VPAD) return;
  int cnt = 0;
  for (int w = lane; w < WROW; w += 32)
    cnt += __popc(adj[(size_t)v * WROW + w]);
  for (int off = 16; off; off >>= 1) cnt += __shfl_down(cnt, off);
  if (lane == 0) {
    float d = (float)cnt;
    degf[v] = d;
    invf[v] = 1.0f / (d + EPS);
  }
}

// x[b,v,d] -> xtc[c*VPAD + v], c = b*3+d (COLUMN-major so B-fragment pairs and
// the finalize's 8-row runs are contiguous). Padded rows stay zero from k_zero.
__global__ void k_transpose(const float* __restrict__ x, float* __restrict__ xtc) {
  int i = blockIdx.x * blockDim.x + threadIdx.x;
  if (i >= VNUM * NCOLS) return;
  int c = i / VNUM, v = i - c * VNUM;
  int b = c / 3, d = c - b * 3;
  xtc[(size_t)c * VPAD + v] = x[((size_t)b * VNUM + v) * 3 + d];
}

// Mark 16x16 blocks (I,J) that contain any adjacency bit.
__global__ void k_blockmap(const unsigned* __restrict__ adj, unsigned* __restrict__ bm) {
  int I = blockIdx.x;
  int J = threadIdx.x;
  if (J >= NT) return;
  unsigned mask = (J & 1) ? 0xFFFF0000u : 0x0000FFFFu;
  int word = J >> 1;                       // cols [J*16, J*16+15] live in word J/2
  unsigned acc = 0;
  for (int m = 0; m < 16; ++m)
    acc |= adj[(size_t)(I * 16 + m) * WROW + word];
  if (acc & mask)
    atomicOr(&bm[(size_t)I * BMW + (J >> 5)], 1u << (J & 31));
}

// Block-sparse SpMM with f32 WMMA. One workgroup per 16-row vertex tile I,
// 6 waves each owning a 16-column chunk of the 96 columns.
// acc = sum_J A(I,J) @ X(J, cols)  with A entries in {-1, 0} (diag excluded).
// Finalize: y = (deg*x + acc) * inv ; partial loss per tile written f32.
__global__ void __launch_bounds__(192) k_gemm(
    const unsigned* __restrict__ adj, const unsigned* __restrict__ bm,
    const float* __restrict__ xtc, const float* __restrict__ degf,
    const float* __restrict__ invf, float* __restrict__ partials) {
  int I = blockIdx.x;
  int tid = threadIdx.x;
  int wave = tid >> 5, lane = tid & 31;

  __shared__ unsigned bmRow[BMW];
  __shared__ float wsum[6];
  if (tid < BMW) bmRow[tid] = bm[(size_t)I * BMW + tid];
  __syncthreads();

  int m    = lane & 15;
  int hi   = lane >> 4;          // 0: lanes 0-15, 1: lanes 16-31
  int kOff = hi << 1;            // A/B fragment K offset: 0 or 2
  int vrow = I * 16 + m;         // A-matrix row handled by this lane
  int n    = wave * 16 + m;      // B/C column handled by this lane
  const unsigned* __restrict__ arow = adj + (size_t)vrow * WROW;
  const float* __restrict__ xcol = xtc + (size_t)n * VPAD;

  v8f acc = {};                  // 16x16 f32 C/D fragment (8 VGPRs)

  for (int jw = 0; jw < BMW; ++jw) {
    unsigned bits = bmRow[jw];   // identical across the wave -> uniform control flow
    while (bits) {
      int j = (jw << 5) + __builtin_ctz(bits);
      bits &= bits - 1;
      // cols [j*16, j*16+15] all live in adjacency word j/2
      unsigned aw = arow[j >> 1];
      // diagonal entries never carry -1 (handled via deg): only possible at j==I
      if (j == I) aw &= ~(1u << (vrow & 31));
      // pre-shift once: bit (4*kk + t) of aws == A[m, kk*4 + kOff + t]
      unsigned aws = aw >> (((j & 1) << 4) + kOff);
      const float* __restrict__ bcol = xcol + (size_t)j * 16 + kOff;
#pragma unroll
      for (int kk = 0; kk < 4; ++kk) {     // K=16 block = 4 x (16x16x4 WMMA)
        // bit -> {0.0f, -1.0f} via sign-extend trick (3 const-operand VALU ops)
        v2f A = {
          __uint_as_float(0xBF800000u &
                          (unsigned)((int)(aws << (31 - (4 * kk + 0))) >> 31)),
          __uint_as_float(0xBF800000u &
                          (unsigned)((int)(aws << (31 - (4 * kk + 1))) >> 31)) };
        // B 4x16 fragment: rows (kOff, kOff+1) are contiguous in column-major xtc
        v2f Bv = *(const v2f*)(bcol + 4 * kk);
        acc = __builtin_amdgcn_wmma_f32_16x16x4_f32(
            /*neg_a=*/false, A, /*neg_b=*/false, Bv,
            /*c_mod=*/(short)0, acc, /*reuse_a=*/false, /*reuse_b=*/false);
      }
    }
  }

  // C/D layout: VGPR g holds (M = g + 8*hi, N = lane&15).
  // No bounds guard needed: padded rows have acc==0, x==0, deg==0 -> y==0.
  int rbase = I * 16 + hi * 8;
  const float* __restrict__ xv = xcol + rbase;   // 8 contiguous floats
  const float* __restrict__ dg = degf + rbase;
  const float* __restrict__ iv = invf + rbase;
  float lsum = 0.0f;
#pragma unroll
  for (int g = 0; g < 8; ++g) {
    float y = (dg[g] * xv[g] + acc[g]) * iv[g];  // acc already = -sum(neighbors)
    lsum += y * y;
  }
  for (int off = 16; off; off >>= 1) lsum += __shfl_down(lsum, off);
  if (lane == 0) wsum[wave] = lsum;
  __syncthreads();
  if (tid == 0) {
    float p = 0.0f;
    for (int w = 0; w < 6; ++w) p += wsum[w];
    partials[I] = p;
  }
}

// Fixed-order scalar reduction -> bitwise-deterministic output.
__global__ void k_reduce(const float* __restrict__ partials, float* __restrict__ out) {
  if (blockIdx.x == 0 && threadIdx.x == 0) {
    float s = 0.0f;
    for (int i = 0; i < NT; ++i) s += partials[i];
    out[0] = s * (1.0f / ((float)BATCH * (float)VNUM));
  }
}

// ---------------- host launcher ----------------
extern "C" void kernel_launch(void* const* d_in, const int* in_sizes, int n_in,
                              void* d_out, int out_size, void* d_ws, size_t ws_size,
                              hipStream_t stream) {
  const float* x     = (const float*)d_in[0];   // [BATCH, VNUM, 3] f32
  const int*   faces = (const int*)d_in[1];     // [NFACES, 3] i32
  float*       out   = (float*)d_out;           // scalar f32

  unsigned* ws   = (unsigned*)d_ws;             // needs ~8.3 MB
  unsigned* adj  = ws + ADJ_OFF;
  float*    degf = (float*)(ws + DEG_OFF);
  float*    invf = (float*)(ws + INV_OFF);
  float*    xtc  = (float*)(ws + XT_OFF);
  unsigned* bm   = ws + BM_OFF;
  float*    part = (float*)(ws + PART_OFF);

  // 1. zero workspace (adj/bm/partials MUST be re-zeroed every call)
  k_zero<<<1024, 256, 0, stream>>>(ws, (long)TOTAL_W);
  // 2. adjacency bits
  k_adj<<<(NFACES + 255) / 256, 256, 0, stream>>>(faces, adj);
  // 3. degrees (one wave per vertex, 8 per block)
  k_deg<<<(VPAD + 7) / 8, 256, 0, stream>>>(adj, degf, invf);
  // 4. x -> column-major [96, VPAD]
  k_transpose<<<(VNUM * NCOLS + 255) / 256, 256, 0, stream>>>(x, xtc);
  // 5. 16x16 block occupancy bitmap
  k_blockmap<<<NT, 448, 0, stream>>>(adj, bm);
  // 6. WMMA block-sparse SpMM + fused loss partials
  k_gemm<<<NT, 192, 0, stream>>>(adj, bm, xtc, degf, invf, part);
  // 7. deterministic final reduction
  k_reduce<<<1, 32, 0, stream>>>(part, out);
}